// _IABlockND_7121055777212
// MI455X (gfx1250) — compile-verified
//
#include <hip/hip_runtime.h>
#include <hip/hip_bf16.h>

#define B_ 128
#define C_ 1024
#define N_ 288            // H*W = 24*12
#define NT_ 18            // N_/16
#define CT_ 64            // C_/16
#define NEG_BIG (-3.402823466e38f)

typedef __attribute__((ext_vector_type(2))) float v2f;
typedef __attribute__((ext_vector_type(8))) float v8f;

__device__ __forceinline__ v8f wmma_f32(v2f a, v2f b, v8f c) {
  // D = A(16x4 f32) * B(4x16 f32) + C(16x16 f32)
  return __builtin_amdgcn_wmma_f32_16x16x4_f32(
      false, a, false, b, (short)0, c, false, false);
}

// ---------------------------------------------------------------------------
// Kernel 1: spatial scores  S[b,p,q] = (1/sqrt(C)) * sum_c x[b,c,p]*x[b,c,q]
// one wave (32 threads) per 16x16 tile of S
// ---------------------------------------------------------------------------
__global__ __launch_bounds__(32) void k_spat_scores(const float* __restrict__ x,
                                                    float* __restrict__ S) {
  const int t  = blockIdx.x;
  const int qt = t % NT_;
  const int pt = (t / NT_) % NT_;
  const int b  = t / (NT_ * NT_);
  const int lane = threadIdx.x;
  const int half = lane >> 4, lid = lane & 15;

  const float* xb = x + (size_t)b * C_ * N_;
  const int colA = pt * 16 + lid;
  const int colB = qt * 16 + lid;

  v8f acc = {};
  for (int k0 = 0; k0 < C_; k0 += 4) {
    const float* xc = xb + (size_t)(k0 + 2 * half) * N_;
    v2f a, bb;
    a.x  = xc[colA];        a.y  = xc[N_ + colA];
    bb.x = xc[colB];        bb.y = xc[N_ + colB];
    acc = wmma_f32(a, bb, acc);
  }
  float* Sb = S + ((size_t)b * N_ + pt * 16) * N_ + qt * 16;
#pragma unroll
  for (int v = 0; v < 8; ++v)
    Sb[(size_t)(v + 8 * half) * N_ + lid] = acc[v] * 0.03125f;  // 1/sqrt(1024)
}

// ---------------------------------------------------------------------------
// Kernel 2: row-wise softmax, * prior, softmax again. One wave per row (len 288)
// ---------------------------------------------------------------------------
__global__ __launch_bounds__(32) void k_softmax2(float* __restrict__ S,
                                                 const float* __restrict__ dis) {
  const int row = blockIdx.x;          // b*N + p
  const int p = row % N_;
  float* r = S + (size_t)row * N_;
  const float* dr = dis + (size_t)p * N_;
  const int lane = threadIdx.x;

  float v[9];
  float m = NEG_BIG;
#pragma unroll
  for (int j = 0; j < 9; ++j) { v[j] = r[lane + 32 * j]; m = fmaxf(m, v[j]); }
  for (int off = 16; off > 0; off >>= 1) m = fmaxf(m, __shfl_xor(m, off, 32));
  float s = 0.f;
#pragma unroll
  for (int j = 0; j < 9; ++j) { v[j] = __expf(v[j] - m); s += v[j]; }
  for (int off = 16; off > 0; off >>= 1) s += __shfl_xor(s, off, 32);
  const float inv = 1.0f / s;

  float m2 = NEG_BIG;
#pragma unroll
  for (int j = 0; j < 9; ++j) {
    v[j] = v[j] * inv * dr[lane + 32 * j];
    m2 = fmaxf(m2, v[j]);
  }
  for (int off = 16; off > 0; off >>= 1) m2 = fmaxf(m2, __shfl_xor(m2, off, 32));
  float s2 = 0.f;
#pragma unroll
  for (int j = 0; j < 9; ++j) { v[j] = __expf(v[j] - m2); s2 += v[j]; }
  for (int off = 16; off > 0; off >>= 1) s2 += __shfl_xor(s2, off, 32);
  const float inv2 = 1.0f / s2;
#pragma unroll
  for (int j = 0; j < 9; ++j) r[lane + 32 * j] = v[j] * inv2;
}

// ---------------------------------------------------------------------------
// Kernel 3: y[b,c,p] = sum_q F[b,p,q] * x[b,c,q]   (K contiguous -> v2f loads)
// ---------------------------------------------------------------------------
__global__ __launch_bounds__(32) void k_spat_apply(const float* __restrict__ x,
                                                   const float* __restrict__ F,
                                                   float* __restrict__ y) {
  const int t  = blockIdx.x;
  const int pt = t % NT_;
  const int ct = (t / NT_) % CT_;
  const int b  = t / (NT_ * CT_);
  const int lane = threadIdx.x;
  const int half = lane >> 4, lid = lane & 15;
  const int c0 = ct * 16, p0 = pt * 16;

  const float* arow = x + (size_t)b * C_ * N_ + (size_t)(c0 + lid) * N_ + 2 * half;
  const float* brow = F + (size_t)b * N_ * N_ + (size_t)(p0 + lid) * N_ + 2 * half;

  v8f acc = {};
  for (int k0 = 0; k0 < N_; k0 += 4) {
    v2f a  = *(const v2f*)(arow + k0);
    v2f bb = *(const v2f*)(brow + k0);
    acc = wmma_f32(a, bb, acc);
  }
#pragma unroll
  for (int v = 0; v < 8; ++v)
    y[((size_t)b * C_ + c0 + v + 8 * half) * N_ + p0 + lid] = acc[v];
}

// ---------------------------------------------------------------------------
// Kernel 4/8: per-channel mean & rsqrt(var+eps) over (B, N)
// ---------------------------------------------------------------------------
__global__ __launch_bounds__(256) void k_bnstats(const float* __restrict__ y,
                                                 float* __restrict__ mean,
                                                 float* __restrict__ rsig) {
  const int c = blockIdx.x;
  const int t = threadIdx.x;
  float s = 0.f, ss = 0.f;
  for (int i = t; i < B_ * N_; i += 256) {
    const int b = i / N_, n = i - b * N_;
    const float val = y[((size_t)b * C_ + c) * N_ + n];
    s += val; ss += val * val;
  }
  __shared__ float sh[256], sh2[256];
  sh[t] = s; sh2[t] = ss;
  __syncthreads();
  for (int o = 128; o > 0; o >>= 1) {
    if (t < o) { sh[t] += sh[t + o]; sh2[t] += sh2[t + o]; }
    __syncthreads();
  }
  if (t == 0) {
    const float cnt = (float)(B_ * N_);
    const float mu = sh[0] / cnt;
    const float var = sh2[0] / cnt - mu * mu;
    mean[c] = mu;
    rsig[c] = rsqrtf(var + 1e-5f);
  }
}

// ---------------------------------------------------------------------------
// Kernel 5/9: out = gamma*(y-mean)*rsig + beta + res   (elementwise)
// ---------------------------------------------------------------------------
__global__ __launch_bounds__(256) void k_bn_add(const float* __restrict__ y,
                                                const float* __restrict__ res,
                                                const float* __restrict__ mean,
                                                const float* __restrict__ rsig,
                                                const float* __restrict__ gamma,
                                                const float* __restrict__ beta,
                                                float* __restrict__ z, long total) {
  for (long i = (long)blockIdx.x * blockDim.x + threadIdx.x; i < total;
       i += (long)gridDim.x * blockDim.x) {
    const int c = (int)((i / N_) % C_);
    z[i] = (y[i] - mean[c]) * rsig[c] * gamma[c] + beta[c] + res[i];
  }
}

// ---------------------------------------------------------------------------
// Kernel 6: channel-attention pass 1 (flash): per-row max & sumexp of
// S2[b,c,d] = sum_n z[b,c,n]*z[b,d,n], without materializing S2 (536 MB saved)
// one wave per (batch, 16-row c-block); online softmax via shuffle reductions
// ---------------------------------------------------------------------------
__global__ __launch_bounds__(32) void k_chan_stats(const float* __restrict__ z,
                                                   float* __restrict__ m2,
                                                   float* __restrict__ l2) {
  const int b  = blockIdx.x >> 6;
  const int c0 = (blockIdx.x & 63) * 16;
  const int lane = threadIdx.x;
  const int half = lane >> 4, lid = lane & 15;
  const float* zb = z + (size_t)b * C_ * N_;
  const float* arow = zb + (size_t)(c0 + lid) * N_ + 2 * half;

  float mrun[8], srun[8];
#pragma unroll
  for (int v = 0; v < 8; ++v) { mrun[v] = NEG_BIG; srun[v] = 0.f; }

  for (int dt = 0; dt < CT_; ++dt) {
    const float* brow = zb + (size_t)(dt * 16 + lid) * N_ + 2 * half;
    v8f sacc = {};
    for (int k0 = 0; k0 < N_; k0 += 4) {
      v2f a  = *(const v2f*)(arow + k0);
      v2f bb = *(const v2f*)(brow + k0);
      sacc = wmma_f32(a, bb, sacc);
    }
#pragma unroll
    for (int v = 0; v < 8; ++v) {
      const float sval = sacc[v];
      float tmax = sval;                       // reduce within 16-lane half
      for (int off = 8; off > 0; off >>= 1)
        tmax = fmaxf(tmax, __shfl_xor(tmax, off, 32));
      const float nm = fmaxf(mrun[v], tmax);
      float tsum = __expf(sval - nm);
      for (int off = 8; off > 0; off >>= 1) tsum += __shfl_xor(tsum, off, 32);
      srun[v] = srun[v] * __expf(mrun[v] - nm) + tsum;
      mrun[v] = nm;
    }
  }
  if (lid == 0) {
#pragma unroll
    for (int v = 0; v < 8; ++v) {
      const int row = c0 + v + 8 * half;
      m2[(size_t)b * C_ + row] = mrun[v];
      l2[(size_t)b * C_ + row] = srun[v];
    }
  }
}

// ---------------------------------------------------------------------------
// Kernel 7: channel-attention pass 2: y2[c,n] = sum_d softmax(S2)[c,d]*z[d,n].
// Recompute each 16x16 score tile once, convert to P with precomputed (m,l),
// stage P through LDS to re-layout C->A fragments, accumulate full n-strip
// (18 x v8f accumulators) in registers.
// ---------------------------------------------------------------------------
__global__ __launch_bounds__(32) void k_chan_apply(const float* __restrict__ z,
                                                   const float* __restrict__ m2,
                                                   const float* __restrict__ l2,
                                                   float* __restrict__ y2) {
  const int b  = blockIdx.x >> 6;
  const int c0 = (blockIdx.x & 63) * 16;
  const int lane = threadIdx.x;
  const int half = lane >> 4, lid = lane & 15;
  const float* zb = z + (size_t)b * C_ * N_;
  __shared__ float Pt[16 * 17];

  float mr[8], il[8];
#pragma unroll
  for (int v = 0; v < 8; ++v) {
    const int row = c0 + v + 8 * half;
    mr[v] = m2[(size_t)b * C_ + row];
    il[v] = 1.0f / l2[(size_t)b * C_ + row];
  }

  v8f acc[NT_];
#pragma unroll
  for (int nt = 0; nt < NT_; ++nt) { v8f zv = {}; acc[nt] = zv; }

  const float* arow = zb + (size_t)(c0 + lid) * N_ + 2 * half;

  for (int dt = 0; dt < CT_; ++dt) {
    const int d0 = dt * 16;
    // --- score tile S2[c0:+16, d0:+16] ---
    const float* brow = zb + (size_t)(d0 + lid) * N_ + 2 * half;
    v8f sacc = {};
    for (int k0 = 0; k0 < N_; k0 += 4) {
      v2f a  = *(const v2f*)(arow + k0);
      v2f bb = *(const v2f*)(brow + k0);
      sacc = wmma_f32(a, bb, sacc);
    }
    // --- P = exp(S - m)/l, staged to LDS for layout conversion ---
#pragma unroll
    for (int v = 0; v < 8; ++v)
      Pt[(v + 8 * half) * 17 + lid] = __expf(sacc[v] - mr[v]) * il[v];
    __syncthreads();
    // --- acc[:, n] += P(16x16) * Z(d0:+16, n) ---
#pragma unroll
    for (int kk = 0; kk < 16; kk += 4) {
      v2f aP;
      aP.x = Pt[lid * 17 + kk + 2 * half];
      aP.y = Pt[lid * 17 + kk + 2 * half + 1];
      const float* zrow0 = zb + (size_t)(d0 + kk + 2 * half) * N_ + lid;
#pragma unroll
      for (int nt = 0; nt < NT_; ++nt) {
        v2f bz;
        bz.x = zrow0[nt * 16];
        bz.y = zrow0[N_ + nt * 16];
        acc[nt] = wmma_f32(aP, bz, acc[nt]);
      }
    }
    __syncthreads();
  }
#pragma unroll
  for (int nt = 0; nt < NT_; ++nt)
#pragma unroll
    for (int v = 0; v < 8; ++v)
      y2[((size_t)b * C_ + c0 + v + 8 * half) * N_ + nt * 16 + lid] = acc[nt][v];
}

// ---------------------------------------------------------------------------
extern "C" void kernel_launch(void* const* d_in, const int* in_sizes, int n_in,
                              void* d_out, int out_size, void* d_ws, size_t ws_size,
                              hipStream_t stream) {
  const float* x      = (const float*)d_in[0];
  const float* dis    = (const float*)d_in[1];
  const float* gamma1 = (const float*)d_in[2];
  const float* beta1  = (const float*)d_in[3];
  const float* gamma2 = (const float*)d_in[4];
  const float* beta2  = (const float*)d_in[5];
  float* out = (float*)d_out;
  float* ws  = (float*)d_ws;

  const size_t BNN = (size_t)B_ * N_ * N_;   // 10.6M floats
  const size_t BCN = (size_t)B_ * C_ * N_;   // 37.7M floats

  float* F     = ws;                 // spatial attention maps
  float* Z     = F + BNN;            // z = BN(y)+x
  float* mean1 = Z + BCN;
  float* rsig1 = mean1 + C_;
  float* mean2 = rsig1 + C_;
  float* rsig2 = mean2 + C_;
  float* m2    = rsig2 + C_;         // B*C row maxes
  float* l2    = m2 + (size_t)B_ * C_;

  // 1) spatial scores (WMMA f32)
  k_spat_scores<<<B_ * NT_ * NT_, 32, 0, stream>>>(x, F);
  // 2) softmax * prior * softmax
  k_softmax2<<<B_ * N_, 32, 0, stream>>>(F, dis);
  // 3) y = F * x^T  (y lives in d_out as scratch)
  k_spat_apply<<<B_ * CT_ * NT_, 32, 0, stream>>>(x, F, out);
  // 4) BN1 stats
  k_bnstats<<<C_, 256, 0, stream>>>(out, mean1, rsig1);
  // 5) z = BN1(y) + x
  k_bn_add<<<4096, 256, 0, stream>>>(out, x, mean1, rsig1, gamma1, beta1, Z, (long)BCN);
  // 6) channel-attention row stats (flash pass 1)
  k_chan_stats<<<B_ * CT_, 32, 0, stream>>>(Z, m2, l2);
  // 7) y2 = softmax(z z^T) z  (flash pass 2, y2 into d_out)
  k_chan_apply<<<B_ * CT_, 32, 0, stream>>>(Z, m2, l2, out);
  // 8) BN2 stats
  k_bnstats<<<C_, 256, 0, stream>>>(out, mean2, rsig2);
  // 9) out = BN2(y2) + z   (in-place elementwise on d_out)
  k_bn_add<<<4096, 256, 0, stream>>>(out, Z, mean2, rsig2, gamma2, beta2, out, (long)BCN);
}